// GAEEnc_73933567033762
// MI455X (gfx1250) — compile-verified
//
#include <hip/hip_runtime.h>

#define N_NODES 100000
#define IN_F    128
#define OUT_F   64
#define N_EDGES 1600000

typedef __attribute__((ext_vector_type(2))) float v2f;
typedef __attribute__((ext_vector_type(8))) float v8f;

__device__ __forceinline__ void atomic_fadd(float* p, float v) {
    // Lowers to GLOBAL_ATOMIC_ADD_F32 (no CAS loop) on AMD
    unsafeAtomicAdd(p, v);
}

// ---- degree / normalization ------------------------------------------------

__global__ void k_init_deg(float* __restrict__ deg) {
    int i = blockIdx.x * blockDim.x + threadIdx.x;
    if (i < N_NODES) deg[i] = 1.0f;  // self-loop contributes 1
}

__global__ void k_degree(const long long* __restrict__ col, float* __restrict__ deg) {
    int e = blockIdx.x * blockDim.x + threadIdx.x;
    if (e < N_EDGES) atomic_fadd(&deg[(int)col[e]], 1.0f);
}

__global__ void k_rsqrt(float* __restrict__ deg) {
    int i = blockIdx.x * blockDim.x + threadIdx.x;
    if (i < N_NODES) deg[i] = rsqrtf(deg[i]);  // deg >= 1 always (self-loops)
}

// ---- h = x @ W via V_WMMA_F32_16X16X4_F32 ----------------------------------
// One wave computes a full 16x64 output stripe (4 accumulators): A fragments
// loaded once per k-step (x read exactly once overall). W is staged in LDS
// K-pair-interleaved: Wp[k/2][col] = {W[k][col], W[k+1][col]}, so every B
// fragment is one contiguous ds_load_b64 into an aligned VGPR pair (no
// register-shuffle movs before the WMMAs).

__global__ void __launch_bounds__(256) k_gemm(const float* __restrict__ x,
                                              const float* __restrict__ W,
                                              float* __restrict__ h) {
    __shared__ float2 Wp[(IN_F / 2) * OUT_F];  // 4096 float2 = 32 KB

    // Cooperative staging: pair (k2, col) <- {W[2*k2][col], W[2*k2+1][col]}
    // Consecutive tids walk consecutive cols -> coalesced global reads.
    const int tid = threadIdx.y * 32 + threadIdx.x;
#pragma unroll
    for (int i = 0; i < (IN_F / 2) * OUT_F / 256; ++i) {   // 16 iters
        const int idx = tid + i * 256;
        const int k2  = idx >> 6;          // 0..63
        const int col = idx & 63;
        float2 p;
        p.x = W[(2 * k2 + 0) * OUT_F + col];
        p.y = W[(2 * k2 + 1) * OUT_F + col];
        Wp[idx] = p;
    }
    __syncthreads();

    const int wave = threadIdx.y;                 // 0..7
    const int lane = threadIdx.x;                 // 0..31
    const int mt   = blockIdx.x * 8 + wave;       // M tile index
    if (mt >= N_NODES / 16) return;               // whole wave exits together
    const int m0 = mt * 16;
    const int lm = lane & 15;                     // M (A) / N (B,D) within tile
    const int lh = lane >> 4;                     // lane-half selects K pair

    const float* __restrict__ arow = x + (size_t)(m0 + lm) * IN_F + lh * 2;

    v8f acc0 = {}, acc1 = {}, acc2 = {}, acc3 = {};
#pragma unroll
    for (int k = 0; k < IN_F; k += 4) {
        // A fragment (ISA 32-bit layout): lane(lm,lh) VGPR v = x[m0+lm][k+2*lh+v]
        const float2 av = *(const float2*)(arow + k);
        v2f a; a.x = av.x; a.y = av.y;
        // B fragments: single b64 LDS load each; VGPR v = W[k+2*lh+v][n0+lm]
        const float2* bp = Wp + ((k >> 1) + lh) * OUT_F + lm;
        const float2 p0 = bp[0];
        const float2 p1 = bp[16];
        const float2 p2 = bp[32];
        const float2 p3 = bp[48];
        v2f b0; b0.x = p0.x; b0.y = p0.y;
        v2f b1; b1.x = p1.x; b1.y = p1.y;
        v2f b2; b2.x = p2.x; b2.y = p2.y;
        v2f b3; b3.x = p3.x; b3.y = p3.y;
        acc0 = __builtin_amdgcn_wmma_f32_16x16x4_f32(false, a, false, b0, (short)0, acc0, false, false);
        acc1 = __builtin_amdgcn_wmma_f32_16x16x4_f32(false, a, false, b1, (short)0, acc1, false, false);
        acc2 = __builtin_amdgcn_wmma_f32_16x16x4_f32(false, a, false, b2, (short)0, acc2, false, false);
        acc3 = __builtin_amdgcn_wmma_f32_16x16x4_f32(false, a, false, b3, (short)0, acc3, false, false);
    }

    // D layout: VGPR v -> row m0 + v + 8*lh, col n0 + lm
    float* hrow = h + (size_t)m0 * OUT_F + lm;
#pragma unroll
    for (int v = 0; v < 8; ++v) {
        const size_t r = (size_t)(v + lh * 8) * OUT_F;
        hrow[r +  0] = acc0[v];
        hrow[r + 16] = acc1[v];
        hrow[r + 32] = acc2[v];
        hrow[r + 48] = acc3[v];
    }
}

// ---- out = b + dis^2 * h  (self-loop term, initializes d_out) --------------

__global__ void k_self_bias(const float* __restrict__ h, const float* __restrict__ dis,
                            const float* __restrict__ b, float* __restrict__ out) {
    size_t idx = (size_t)blockIdx.x * blockDim.x + threadIdx.x;
    if (idx >= (size_t)N_NODES * OUT_F) return;
    int i = (int)(idx >> 6);
    int f = (int)(idx & 63);
    float d = dis[i];
    out[idx] = b[f] + d * d * h[idx];
}

// ---- edge scatter: out[col] += dis[row]*dis[col] * h[row] ------------------
// One wave32 per edge; each lane: float2 gather + two fp32 global atomics.
// h/out/dis are L2-resident (51 MB total << 192 MB), so this phase is bounded
// by L2 atomic throughput, not HBM.

__global__ void __launch_bounds__(256) k_scatter(const long long* __restrict__ ei,
                                                 const float* __restrict__ h,
                                                 const float* __restrict__ dis,
                                                 float* __restrict__ out) {
    const size_t gid  = (size_t)blockIdx.x * blockDim.x + threadIdx.x;
    const size_t e    = gid >> 5;
    const int    lane = (int)(gid & 31);
    if (e >= N_EDGES) return;
    const int   r    = (int)ei[e];
    const int   c    = (int)ei[(size_t)N_EDGES + e];
    const float norm = dis[r] * dis[c];
    const float2 hv  = *(const float2*)(h + (size_t)r * OUT_F + lane * 2);
    float* op = out + (size_t)c * OUT_F + lane * 2;
    atomic_fadd(op,     norm * hv.x);
    atomic_fadd(op + 1, norm * hv.y);
}

// ---- driver ----------------------------------------------------------------

extern "C" void kernel_launch(void* const* d_in, const int* in_sizes, int n_in,
                              void* d_out, int out_size, void* d_ws, size_t ws_size,
                              hipStream_t stream) {
    const float*     x  = (const float*)d_in[0];
    const long long* ei = (const long long*)d_in[1];  // int64 edge_index [2, E]
    const float*     W  = (const float*)d_in[2];
    const float*     b  = (const float*)d_in[3];
    float* out = (float*)d_out;

    float* dis = (float*)d_ws;            // N_NODES floats (400000 B, 16B aligned)
    float* h   = dis + N_NODES;           // N_NODES*OUT_F floats (25.6 MB)

    k_init_deg<<<(N_NODES + 255) / 256, 256, 0, stream>>>(dis);
    k_degree<<<(N_EDGES + 255) / 256, 256, 0, stream>>>(ei + N_EDGES, dis);
    k_rsqrt<<<(N_NODES + 255) / 256, 256, 0, stream>>>(dis);

    dim3 gblk(32, 8);
    const int m_tiles = N_NODES / 16;                   // 6250
    k_gemm<<<(m_tiles + 7) / 8, gblk, 0, stream>>>(x, W, h);

    k_self_bias<<<((size_t)N_NODES * OUT_F + 255) / 256, 256, 0, stream>>>(h, dis, b, out);

    k_scatter<<<(unsigned)(((size_t)N_EDGES * 32 + 255) / 256), 256, 0, stream>>>(ei, h, dis, out);
}